// GIN_17162689314898
// MI455X (gfx1250) — compile-verified
//
#include <hip/hip_runtime.h>
#include <math.h>

#define BN_EPS 1e-5f

typedef __attribute__((ext_vector_type(2))) float v2f;
typedef __attribute__((ext_vector_type(8))) float v8f;

// ---------------------------------------------------------------- fills
__global__ void fill_f32_kernel(float* __restrict__ p, float v, int n) {
  int i = blockIdx.x * blockDim.x + threadIdx.x;
  if (i < n) p[i] = v;
}
__global__ void fill_i32_kernel(int* __restrict__ p, int v, int n) {
  int i = blockIdx.x * blockDim.x + threadIdx.x;
  if (i < n) p[i] = v;
}

// ---------------------------------------------------------------- edge scatter
// agg[dst] += h[src]  (GIN sum aggregation). Streaming, atomics-bound.
__global__ void agg_edges_kernel(const float* __restrict__ h,
                                 const int* __restrict__ src,
                                 const int* __restrict__ dst,
                                 float* __restrict__ agg, int E) {
  int e = blockIdx.x * blockDim.x + threadIdx.x;
  if (e >= E) return;
  // gfx1250 global_prefetch_b8: pull index stream ahead (speculative, OOB-safe)
  __builtin_prefetch(&src[e + 4096], 0, 1);
  __builtin_prefetch(&dst[e + 4096], 0, 1);
  int s = src[e];
  int d = dst[e];
  float2 v = *reinterpret_cast<const float2*>(h + 2u * (unsigned)s);
  atomicAdd(&agg[2u * (unsigned)d + 0], v.x);
  atomicAdd(&agg[2u * (unsigned)d + 1], v.y);
}

// ---------------------------------------------------------------- conv1 pass A
// z = (x + agg) @ w1 + b1 ; accumulate sum/sumsq per channel for BN.
__global__ void conv1_z_kernel(const float* __restrict__ x,
                               const float* __restrict__ agg,
                               const float* __restrict__ w1,   // [2,2] row-major
                               const float* __restrict__ b1,   // [2]
                               float* __restrict__ z,
                               float* __restrict__ bnacc,      // [4]: s0,s1,q0,q1
                               int N) {
  __shared__ float s0[256], s1[256], s2[256], s3[256];
  int t = threadIdx.x;
  int i = blockIdx.x * blockDim.x + t;
  float z0 = 0.f, z1 = 0.f;
  if (i < N) {
    float gx = x[2 * i + 0] + agg[2 * i + 0];
    float gy = x[2 * i + 1] + agg[2 * i + 1];
    z0 = gx * w1[0] + gy * w1[2] + b1[0];
    z1 = gx * w1[1] + gy * w1[3] + b1[1];
    z[2 * i + 0] = z0;
    z[2 * i + 1] = z1;
  }
  s0[t] = z0; s1[t] = z1; s2[t] = z0 * z0; s3[t] = z1 * z1;
  __syncthreads();
  for (int off = 128; off > 0; off >>= 1) {
    if (t < off) {
      s0[t] += s0[t + off]; s1[t] += s1[t + off];
      s2[t] += s2[t + off]; s3[t] += s3[t + off];
    }
    __syncthreads();
  }
  if (t == 0) {
    atomicAdd(&bnacc[0], s0[0]);
    atomicAdd(&bnacc[1], s1[0]);
    atomicAdd(&bnacc[2], s2[0]);
    atomicAdd(&bnacc[3], s3[0]);
  }
}

// ---------------------------------------------------------------- conv1 pass B
// zn = BN(z); h = relu(relu(zn) @ w2 + b2)
__global__ void conv1_apply_kernel(const float* __restrict__ z,
                                   const float* __restrict__ bnacc,
                                   const float* __restrict__ g,
                                   const float* __restrict__ beta,
                                   const float* __restrict__ w2,  // [2,2]
                                   const float* __restrict__ b2,  // [2]
                                   float* __restrict__ h, int N) {
  int i = blockIdx.x * blockDim.x + threadIdx.x;
  if (i >= N) return;
  float inv_n = 1.f / (float)N;
  float mu0 = bnacc[0] * inv_n, mu1 = bnacc[1] * inv_n;
  float v0 = bnacc[2] * inv_n - mu0 * mu0;
  float v1 = bnacc[3] * inv_n - mu1 * mu1;
  float is0 = rsqrtf(v0 + BN_EPS), is1 = rsqrtf(v1 + BN_EPS);
  float zn0 = (z[2 * i + 0] - mu0) * is0 * g[0] + beta[0];
  float zn1 = (z[2 * i + 1] - mu1) * is1 * g[1] + beta[1];
  float r0 = zn0 > 0.f ? zn0 : 0.f;
  float r1 = zn1 > 0.f ? zn1 : 0.f;
  float h0 = r0 * w2[0] + r1 * w2[2] + b2[0];
  float h1 = r0 * w2[1] + r1 * w2[3] + b2[1];
  h[2 * i + 0] = h0 > 0.f ? h0 : 0.f;
  h[2 * i + 1] = h1 > 0.f ? h1 : 0.f;
}

// ---------------------------------------------------------------- conv2..7 node op
// h_out = relu(relu((h_in + agg) @ w1 + b1) @ w2 + b2)
__global__ void conv_apply_kernel(const float* __restrict__ h_in,
                                  const float* __restrict__ agg,
                                  const float* __restrict__ w1,
                                  const float* __restrict__ b1,
                                  const float* __restrict__ w2,
                                  const float* __restrict__ b2,
                                  float* __restrict__ h_out, int N) {
  int i = blockIdx.x * blockDim.x + threadIdx.x;
  if (i >= N) return;
  float gx = h_in[2 * i + 0] + agg[2 * i + 0];
  float gy = h_in[2 * i + 1] + agg[2 * i + 1];
  float z0 = gx * w1[0] + gy * w1[2] + b1[0];
  float z1 = gx * w1[1] + gy * w1[3] + b1[1];
  z0 = z0 > 0.f ? z0 : 0.f;
  z1 = z1 > 0.f ? z1 : 0.f;
  float o0 = z0 * w2[0] + z1 * w2[2] + b2[0];
  float o1 = z0 * w2[1] + z1 * w2[3] + b2[1];
  h_out[2 * i + 0] = o0 > 0.f ? o0 : 0.f;
  h_out[2 * i + 1] = o1 > 0.f ? o1 : 0.f;
}

// ---------------------------------------------------------------- graph pooling
// adds[l][g][c] += h_l[i][c] ; maxs = max (int-punned; values >= 0 post-ReLU)
__global__ void pool_kernel(const float* __restrict__ hs,  // 7 x [N,2] contiguous
                            const int* __restrict__ batch,
                            float* __restrict__ adds,      // [7,G,2]
                            int* __restrict__ maxs,        // [7,G,2]
                            int N, int G) {
  int i = blockIdx.x * blockDim.x + threadIdx.x;
  if (i >= N) return;
  int b = batch[i];
  for (int l = 0; l < 7; ++l) {
    float a0 = hs[(size_t)l * 2 * N + 2 * i + 0];
    float a1 = hs[(size_t)l * 2 * N + 2 * i + 1];
    int base = (l * G + b) * 2;
    atomicAdd(&adds[base + 0], a0);
    atomicAdd(&adds[base + 1], a1);
    atomicMax(&maxs[base + 0], __float_as_int(a0));
    atomicMax(&maxs[base + 1], __float_as_int(a1));
  }
}

// pooled[g, 0:14] = adds, pooled[g, 14:28] = maxs
__global__ void build_pooled_kernel(const float* __restrict__ adds,
                                    const int* __restrict__ maxs,
                                    float* __restrict__ pooled, int G) {
  int idx = blockIdx.x * blockDim.x + threadIdx.x;
  if (idx >= G * 28) return;
  int g = idx / 28, j = idx % 28;
  float v;
  if (j < 14) {
    v = adds[((j >> 1) * G + g) * 2 + (j & 1)];
  } else {
    int jj = j - 14;
    v = __int_as_float(maxs[((jj >> 1) * G + g) * 2 + (jj & 1)]);
  }
  pooled[g * 28 + j] = v;
}

// ---------------------------------------------------------------- readout GEMM
// h_read = relu(pooled[G,28] @ W[28,28] + b) via V_WMMA_F32_16X16X4_F32.
// One wave32 per 16x16 output tile; K=28 = 7 chained x4-K WMMAs.
// A layout (ISA 7.12.2, 32-bit A 16x4): lane<16 -> M=lane, K={0,1}+4k;
//                                       lane>=16 -> M=lane-16, K={2,3}+4k.
// B layout: lanes 0-15 hold lower K pair, lanes 16-31 upper pair, VGPR steps K.
//
// No predication around the WMMAs: out-of-range rows/cols use CLAMPED addresses
// (always in-bounds loads, EXEC stays all-ones). Garbage values in A row m /
// B col n only affect output row m / col n, which are masked at the store.
__global__ __launch_bounds__(32) void readout_wmma_kernel(
    const float* __restrict__ pooled, const float* __restrict__ W,
    const float* __restrict__ bias, float* __restrict__ h_read, int G) {
  const int lane = threadIdx.x;
  const int tm = blockIdx.x;      // M tile
  const int tn = blockIdx.y;      // N tile (0 or 1; cols 16..27 valid in tile 1)
  const int half = lane >> 4;     // 0 | 1
  const int l16 = lane & 15;

  int arow = tm * 16 + l16;
  if (arow > G - 1) arow = G - 1;          // clamp: keep loads in-bounds
  int bcol = tn * 16 + l16;
  if (bcol > 27) bcol = 27;                // clamp: keep loads in-bounds

  const float* arow_p = pooled + arow * 28;

  v8f c = {};
#pragma unroll
  for (int kk = 0; kk < 7; ++kk) {
    const int kb = kk * 4 + half * 2;      // K base for this lane half
    // A pair is contiguous + 8B aligned (row*28 even, kb even) -> one b64 load
    v2f a = *reinterpret_cast<const v2f*>(arow_p + kb);
    v2f b;
    b.x = W[(kb + 0) * 28 + bcol];
    b.y = W[(kb + 1) * 28 + bcol];
    c = __builtin_amdgcn_wmma_f32_16x16x4_f32(
        /*neg_a=*/false, a, /*neg_b=*/false, b,
        /*c_mod=*/(short)0, c, /*reuse_a=*/false, /*reuse_b=*/false);
  }
  // C/D layout: VGPR v, lanes 0-15 -> M=v, lanes 16-31 -> M=v+8
#pragma unroll
  for (int v = 0; v < 8; ++v) {
    int r = tm * 16 + v + half * 8;
    int col = tn * 16 + l16;
    if (r < G && col < 28) {
      float val = c[v] + bias[col];
      h_read[r * 28 + col] = val > 0.f ? val : 0.f;
    }
  }
}

// ---------------------------------------------------------------- head
__global__ void head_kernel(const float* __restrict__ h_read,
                            const float* __restrict__ w2,  // [28]
                            const float* __restrict__ b2,  // [1]
                            float* __restrict__ out, int G) {
  int g = blockIdx.x * blockDim.x + threadIdx.x;
  if (g >= G) return;
  float acc = b2[0];
  for (int j = 0; j < 28; ++j) acc += h_read[g * 28 + j] * w2[j];
  out[g] = 1.f / (1.f + expf(-acc));  // sigmoid(logits)
  out[G + g] = acc;                   // logits
}

// ================================================================ launch
extern "C" void kernel_launch(void* const* d_in, const int* in_sizes, int n_in,
                              void* d_out, int out_size, void* d_ws, size_t ws_size,
                              hipStream_t stream) {
  const float* x      = (const float*)d_in[0];
  const float* c1_w1  = (const float*)d_in[1];
  const float* c1_b1  = (const float*)d_in[2];
  const float* bn_g   = (const float*)d_in[3];
  const float* bn_b   = (const float*)d_in[4];
  const float* c1_w2  = (const float*)d_in[5];
  const float* c1_b2  = (const float*)d_in[6];
  const float* cw1    = (const float*)d_in[7];   // [6,2,2]
  const float* cb1    = (const float*)d_in[8];   // [6,2]
  const float* cw2    = (const float*)d_in[9];   // [6,2,2]
  const float* cb2    = (const float*)d_in[10];  // [6,2]
  const float* lin1_w = (const float*)d_in[11];  // [28,28]
  const float* lin1_b = (const float*)d_in[12];  // [28]
  const float* lin2_w = (const float*)d_in[13];  // [28,1]
  const float* lin2_b = (const float*)d_in[14];  // [1]
  const int*   eidx   = (const int*)d_in[15];    // [2,E]
  const int*   batch  = (const int*)d_in[16];    // [N]

  const int N = in_sizes[16];
  const int E = in_sizes[15] / 2;
  const int G = out_size / 2;
  const int* src = eidx;
  const int* dst = eidx + E;

  // ---- workspace carve-up (256B aligned slices)
  char* ws = (char*)d_ws;
  size_t off = 0;
  auto carve = [&](size_t bytes) -> void* {
    off = (off + 255) & ~(size_t)255;
    void* p = ws + off;
    off += bytes;
    return p;
  };
  float* agg    = (float*)carve(sizeof(float) * 2 * (size_t)N);
  float* zbuf   = (float*)carve(sizeof(float) * 2 * (size_t)N);
  float* hs     = (float*)carve(sizeof(float) * 2 * (size_t)N * 7);
  float* bnacc  = (float*)carve(sizeof(float) * 4);
  float* adds   = (float*)carve(sizeof(float) * 14 * (size_t)G);
  int*   maxs   = (int*)  carve(sizeof(int)   * 14 * (size_t)G);
  float* pooled = (float*)carve(sizeof(float) * 28 * (size_t)G);
  float* hread  = (float*)carve(sizeof(float) * 28 * (size_t)G);
  (void)ws_size;

  const int B = 256;
  const int gN  = (N + B - 1) / B;
  const int gE  = (E + B - 1) / B;
  const int g2N = (2 * N + B - 1) / B;
  const int gP  = (14 * G + B - 1) / B;

  // bit pattern of -INFINITY (host-side constant; __float_as_int is device-only)
  const int NEG_INF_BITS = (int)0xFF800000u;

  // ---- conv1
  fill_f32_kernel<<<g2N, B, 0, stream>>>(agg, 0.f, 2 * N);
  fill_f32_kernel<<<1, 32, 0, stream>>>(bnacc, 0.f, 4);
  agg_edges_kernel<<<gE, B, 0, stream>>>(x, src, dst, agg, E);
  conv1_z_kernel<<<gN, B, 0, stream>>>(x, agg, c1_w1, c1_b1, zbuf, bnacc, N);
  conv1_apply_kernel<<<gN, B, 0, stream>>>(zbuf, bnacc, bn_g, bn_b, c1_w2, c1_b2,
                                           hs /* layer 0 */, N);

  // ---- conv2..conv7
  for (int i = 0; i < 6; ++i) {
    float* h_in  = hs + (size_t)i * 2 * N;
    float* h_out = hs + (size_t)(i + 1) * 2 * N;
    fill_f32_kernel<<<g2N, B, 0, stream>>>(agg, 0.f, 2 * N);
    agg_edges_kernel<<<gE, B, 0, stream>>>(h_in, src, dst, agg, E);
    conv_apply_kernel<<<gN, B, 0, stream>>>(h_in, agg, cw1 + i * 4, cb1 + i * 2,
                                            cw2 + i * 4, cb2 + i * 2, h_out, N);
  }

  // ---- pooling (sum + max per graph, 7 layers)
  fill_f32_kernel<<<gP, B, 0, stream>>>(adds, 0.f, 14 * G);
  fill_i32_kernel<<<gP, B, 0, stream>>>(maxs, NEG_INF_BITS, 14 * G);
  pool_kernel<<<gN, B, 0, stream>>>(hs, batch, adds, maxs, N, G);
  build_pooled_kernel<<<(G * 28 + B - 1) / B, B, 0, stream>>>(adds, maxs, pooled, G);

  // ---- readout GEMM on the matrix engine (wave32 WMMA tiles)
  dim3 wgrid((G + 15) / 16, 2);
  readout_wmma_kernel<<<wgrid, 32, 0, stream>>>(pooled, lin1_w, lin1_b, hread, G);

  // ---- head: logits + sigmoid
  head_kernel<<<(G + B - 1) / B, B, 0, stream>>>(hread, lin2_w, lin2_b,
                                                 (float*)d_out, G);
}